// GridEncoderMinkowski_10754598109276
// MI455X (gfx1250) — compile-verified
//
#include <hip/hip_runtime.h>
#include <stdint.h>

#define LLEVELS 16
#define SHIFT   21
#define OFFC    (1ll << 20)
#define NS      8192        // LDS key samples per level (64 KB of LDS, stride 64)
#define TPB     256
#define PPT     8           // points per thread
#define PPB     (TPB * PPT) // 2048 points per block

__global__ __launch_bounds__(TPB)
void grid_encode_kernel(const float* __restrict__ inputs,
                        const float* __restrict__ emb,
                        const long long* __restrict__ keys,
                        const int* __restrict__ resolutions,
                        float* __restrict__ out,
                        int M, int N)
{
    const int l = blockIdx.y;
    const long long* __restrict__ kl = keys + (size_t)l * (size_t)N;
    const float2*    __restrict__ el = (const float2*)(emb + (size_t)l * (size_t)N * 2u);

    __shared__ long long smp[NS];

    const int  S      = N / NS;                 // sample stride (64 for N = 2^19)
    const bool useIdx = (S > 0) && (S * NS == N);

    // ---- Stage every S-th key into LDS via gfx1250 async global->LDS copies ----
    if (useIdx) {
        for (int k = threadIdx.x; k < NS; k += TPB) {
            const long long* gsrc = kl + (size_t)k * (size_t)S;
            // low 32 bits of the flat address of an LDS object == LDS byte offset
            unsigned lds_off = (unsigned)(uintptr_t)(&smp[k]);
            asm volatile("global_load_async_to_lds_b64 %0, %1, off"
                         :: "v"(lds_off), "v"(gsrc)
                         : "memory");
        }
        asm volatile("s_wait_asynccnt 0" ::: "memory");
        __builtin_amdgcn_s_wait_tensorcnt(0);   // no TDM outstanding; cheap fence on new counter
    }
    __syncthreads();

    const float scale = (float)resolutions[l] * 0.5f;   // res / (bound * 2), bound = 1
    const int base_i  = blockIdx.x * PPB + threadIdx.x;

    #pragma unroll 1
    for (int p = 0; p < PPT; ++p) {
        const int i = base_i + p * TPB;
        if (i >= M) continue;

        // reference permutes columns: x' = (in[2], in[0], in[1])
        const float c0 = inputs[3 * i + 2];
        const float c1 = inputs[3 * i + 0];
        const float c2 = inputs[3 * i + 1];

        const float q0 = c0 * scale, q1 = c1 * scale, q2 = c2 * scale;
        const float f0 = floorf(q0), f1 = floorf(q1), f2 = floorf(q2);
        const float r0 = q0 - f0,    r1 = q1 - f1,    r2 = q2 - f2;

        const long long b0 = (long long)f0 + OFFC;
        const long long b1 = (long long)f1 + OFFC;
        const long long b2 = (long long)f2 + OFFC;
        const long long key00 = (b0 << (2 * SHIFT)) + (b1 << SHIFT) + b2;

        const float wx[2] = {1.0f - r0, r0};
        const float wy[2] = {1.0f - r1, r1};
        const float wz0 = 1.0f - r2, wz1 = r2;

        // 4 independent search chains: (dx,dy); dz handled analytically (key +1)
        long long k4[4];
        int bs[4], ln[4];
        #pragma unroll
        for (int j = 0; j < 4; ++j) {
            const int dx = j >> 1, dy = j & 1;
            k4[j] = key00 + ((long long)dx << (2 * SHIFT)) + ((long long)dy << SHIFT);
        }

        if (useIdx) {
            // branchless lower_bound over LDS samples (NS is a power of two): 13 LDS steps
            int sj[4] = {0, 0, 0, 0};
            for (int len = NS; len > 1;) {
                const int half = len >> 1;
                #pragma unroll
                for (int j = 0; j < 4; ++j) {
                    const long long v = smp[sj[j] + half - 1];
                    sj[j] += (v < k4[j]) ? half : 0;
                }
                len -= half;
            }
            #pragma unroll
            for (int j = 0; j < 4; ++j) {
                const int s0 = sj[j] + ((smp[sj[j]] < k4[j]) ? 1 : 0); // in [0, NS]
                const int lo = (s0 == 0) ? 0 : (s0 - 1) * S;
                const int hi = (s0 == NS) ? N : (s0 * S + 1);
                bs[j] = lo;
                ln[j] = hi - lo;      // <= S + 1 (65): ~7 more L2-resident steps
            }
        } else {
            #pragma unroll
            for (int j = 0; j < 4; ++j) { bs[j] = 0; ln[j] = N; }
        }

        // lockstep branchless lower_bound over global keys: 4 loads in flight per step
        bool any = true;
        while (any) {
            any = false;
            #pragma unroll
            for (int j = 0; j < 4; ++j) {
                if (ln[j] > 1) {
                    const int half = ln[j] >> 1;
                    const long long v = kl[bs[j] + half - 1];
                    if (v < k4[j]) bs[j] += half;
                    ln[j] -= half;
                    if (ln[j] > 1) any = true;
                }
            }
        }

        float acc0 = 0.0f, acc1 = 0.0f;
        #pragma unroll
        for (int j = 0; j < 4; ++j) {
            const int dx = j >> 1, dy = j & 1;
            const float wxy = wx[dx] * wy[dy];

            const int lb = bs[j] + ((kl[bs[j]] < k4[j]) ? 1 : 0);  // lower_bound in [0, N]

            // dz = 0
            const int       i0  = min(lb, N - 1);
            const long long kv0 = kl[i0];
            const float2    e0  = el[i0];
            const float     h0  = (kv0 == k4[j]) ? 1.0f : 0.0f;
            const float     w0  = wxy * wz0 * h0;
            acc0 += w0 * e0.x;
            acc1 += w0 * e0.y;

            // dz = 1: keys unique+sorted => lb(key+1) = lb + (keys[lb] == key)
            const int       lb1 = lb + (((lb < N) && (kv0 == k4[j])) ? 1 : 0);
            const int       i1  = min(lb1, N - 1);
            const long long kv1 = kl[i1];
            const float2    e1  = el[i1];
            const float     h1  = (kv1 == k4[j] + 1) ? 1.0f : 0.0f;
            const float     w1  = wxy * wz1 * h1;
            acc0 += w1 * e1.x;
            acc1 += w1 * e1.y;
        }

        // streaming output: non-temporal so the 128 MB result doesn't evict the
        // L2-resident key/emb tables (~128 MB of 192 MB L2)
        float* op = out + (size_t)i * (2 * LLEVELS) + (size_t)l * 2;
        __builtin_nontemporal_store(acc0, op + 0);
        __builtin_nontemporal_store(acc1, op + 1);
    }
}

extern "C" void kernel_launch(void* const* d_in, const int* in_sizes, int n_in,
                              void* d_out, int out_size, void* d_ws, size_t ws_size,
                              hipStream_t stream) {
    const float*     inputs      = (const float*)d_in[0];
    const float*     emb         = (const float*)d_in[1];
    const long long* keys        = (const long long*)d_in[2];
    const int*       resolutions = (const int*)d_in[3];
    float*           out         = (float*)d_out;

    const int M = in_sizes[0] / 3;
    const int N = in_sizes[2] / LLEVELS;

    dim3 grid((M + PPB - 1) / PPB, LLEVELS);
    grid_encode_kernel<<<grid, TPB, 0, stream>>>(inputs, emb, keys, resolutions, out, M, N);
}